// GCN_6871947674334
// MI455X (gfx1250) — compile-verified
//
#include <hip/hip_runtime.h>

typedef __attribute__((ext_vector_type(2))) float v2f;
typedef __attribute__((ext_vector_type(8))) float v8f;

#define F_IN  165
#define F_HID 128
#define F_OUT 2

// ---------------------------------------------------------------- degrees ---
__global__ void deg_init_kernel(float* __restrict__ deg, int n) {
    int i = blockIdx.x * blockDim.x + threadIdx.x;
    if (i < n) deg[i] = 1.0f;                       // self-loop contribution
}

__global__ void deg_count_kernel(const long long* __restrict__ dst,
                                 float* __restrict__ deg, long long E) {
    long long e = (long long)blockIdx.x * blockDim.x + threadIdx.x;
    if (e < E) atomicAdd(&deg[dst[e]], 1.0f);
}

__global__ void deg_rsqrt_kernel(float* __restrict__ deg, int n) {
    int i = blockIdx.x * blockDim.x + threadIdx.x;
    if (i < n) deg[i] = rsqrtf(deg[i]);             // deg >= 1 always
}

// ------------------------------------------------- GEMM1: h = x @ W1 (WMMA) --
// One wave computes a 16x16 tile of h. 8 waves/block cover all 8 col-tiles of
// one 16-row stripe. fp32 WMMA 16x16x4, K tail (165 = 41*4 + 1) zero-padded
// via per-lane selects so EXEC stays all-ones around v_wmma.
__global__ __launch_bounds__(256) void gemm1_wmma_kernel(
        const float* __restrict__ x, const float* __restrict__ W1,
        float* __restrict__ h, int n) {
    const int lane    = threadIdx.x & 31;
    const int wave    = threadIdx.x >> 5;          // 0..7 -> column tile
    const int rowTile = blockIdx.x;
    const int half    = lane >> 4;                 // 0: lanes 0-15, 1: 16-31
    const int l15     = lane & 15;
    const int colBase = wave * 16;
    const int kOff    = half * 2;                  // A/B VGPR K placement

    int arow = rowTile * 16 + l15;
    if (arow >= n) arow = n - 1;                   // clamp (store is guarded)
    const float* xr = x + (size_t)arow * F_IN;

    v8f c = {0.f, 0.f, 0.f, 0.f, 0.f, 0.f, 0.f, 0.f};
    #pragma unroll 4
    for (int k = 0; k < 168; k += 4) {
        const int k0 = k + kOff, k1 = k0 + 1;
        v2f a, b;
        a.x = (k0 < F_IN) ? xr[k0] : 0.0f;
        a.y = (k1 < F_IN) ? xr[k1] : 0.0f;
        b.x = (k0 < F_IN) ? W1[(size_t)k0 * F_HID + colBase + l15] : 0.0f;
        b.y = (k1 < F_IN) ? W1[(size_t)k1 * F_HID + colBase + l15] : 0.0f;
        c = __builtin_amdgcn_wmma_f32_16x16x4_f32(
                /*neg_a=*/false, a, /*neg_b=*/false, b,
                /*c_mod=*/(short)0, c, /*reuse_a=*/false, /*reuse_b=*/false);
    }
    // C/D layout: VGPR v -> M = v + 8*half, N = l15
    const int rbase = rowTile * 16 + half * 8;
    #pragma unroll
    for (int v = 0; v < 8; ++v) {
        int r = rbase + v;
        if (r < n) h[(size_t)r * F_HID + colBase + l15] = c[v];
    }
}

// ---------------------------------------------------- layer-1 aggregation ---
__global__ void agg1_init_kernel(const float* __restrict__ h,
                                 const float* __restrict__ dinv,
                                 float* __restrict__ out1, int n) {
    int idx = blockIdx.x * blockDim.x + threadIdx.x;   // n * 128 elements
    if (idx < n * F_HID) {
        float d = dinv[idx >> 7];                      // /128
        out1[idx] = d * d * h[idx];                    // self-loop message
    }
}

// one wave per edge: 32 lanes x float4 = 128 features, coalesced
__global__ __launch_bounds__(256) void agg1_edge_kernel(
        const long long* __restrict__ src, const long long* __restrict__ dst,
        const float* __restrict__ dinv, const float* __restrict__ h,
        float* __restrict__ out1, long long E) {
    long long e = (long long)blockIdx.x * 8 + (threadIdx.x >> 5);
    if (e >= E) return;
    const int lane = threadIdx.x & 31;
    const long long s = src[e], d = dst[e];
    const float norm = dinv[s] * dinv[d];
    const float4 v = ((const float4*)(h + (size_t)s * F_HID))[lane];
    float* o = out1 + (size_t)d * F_HID + lane * 4;
    atomicAdd(o + 0, norm * v.x);
    atomicAdd(o + 1, norm * v.y);
    atomicAdd(o + 2, norm * v.z);
    atomicAdd(o + 3, norm * v.w);
}

// ------------------------------- fused bias + ReLU + GEMM2 (128 -> 2) -------
__global__ void gemm2_kernel(const float* __restrict__ out1,
                             const float* __restrict__ b1,
                             const float* __restrict__ W2,
                             float* __restrict__ h2, int n) {
    int i = blockIdx.x * blockDim.x + threadIdx.x;
    if (i >= n) return;
    const float* r = out1 + (size_t)i * F_HID;
    float a0 = 0.f, a1 = 0.f;
    #pragma unroll 8
    for (int c = 0; c < F_HID; ++c) {
        float a = r[c] + b1[c];
        a = a > 0.f ? a : 0.f;                         // ReLU
        a0 += a * W2[c * 2 + 0];
        a1 += a * W2[c * 2 + 1];
    }
    h2[i * 2 + 0] = a0;
    h2[i * 2 + 1] = a1;
}

// ---------------------------------------------------- layer-2 aggregation ---
__global__ void agg2_init_kernel(const float* __restrict__ h2,
                                 const float* __restrict__ dinv,
                                 const float* __restrict__ b2,
                                 float* __restrict__ out, int n) {
    int i = blockIdx.x * blockDim.x + threadIdx.x;
    if (i >= n) return;
    float d = dinv[i], dd = d * d;
    out[i * 2 + 0] = dd * h2[i * 2 + 0] + b2[0];
    out[i * 2 + 1] = dd * h2[i * 2 + 1] + b2[1];
}

__global__ void agg2_edge_kernel(const long long* __restrict__ src,
                                 const long long* __restrict__ dst,
                                 const float* __restrict__ dinv,
                                 const float* __restrict__ h2,
                                 float* __restrict__ out, long long E) {
    long long e = (long long)blockIdx.x * blockDim.x + threadIdx.x;
    if (e >= E) return;
    const long long s = src[e], d = dst[e];
    const float norm = dinv[s] * dinv[d];
    atomicAdd(&out[d * 2 + 0], norm * h2[s * 2 + 0]);
    atomicAdd(&out[d * 2 + 1], norm * h2[s * 2 + 1]);
}

// ---------------------------------------------------------------- launcher --
extern "C" void kernel_launch(void* const* d_in, const int* in_sizes, int n_in,
                              void* d_out, int out_size, void* d_ws, size_t ws_size,
                              hipStream_t stream) {
    const float*     x  = (const float*)d_in[0];
    const long long* ei = (const long long*)d_in[1];   // int64 edge_index
    const float*     W1 = (const float*)d_in[2];
    const float*     b1 = (const float*)d_in[3];
    const float*     W2 = (const float*)d_in[4];
    const float*     b2 = (const float*)d_in[5];
    float*           out = (float*)d_out;

    const int       N = in_sizes[0] / F_IN;            // 100000
    const long long E = (long long)in_sizes[1] / 2;    // 1600000
    const long long* src = ei;
    const long long* dst = ei + E;

    // workspace carve-up (floats)
    float* ws   = (float*)d_ws;
    float* dinv = ws;                                  // N
    float* h    = ws + (((size_t)N + 255) & ~(size_t)255);
    float* out1 = h + (size_t)N * F_HID;               // N*128
    float* h2   = out1 + (size_t)N * F_HID;            // N*2

    // 1) symmetric-normalization coefficients
    deg_init_kernel<<<(N + 255) / 256, 256, 0, stream>>>(dinv, N);
    deg_count_kernel<<<(int)((E + 255) / 256), 256, 0, stream>>>(dst, dinv, E);
    deg_rsqrt_kernel<<<(N + 255) / 256, 256, 0, stream>>>(dinv, N);

    // 2) h = x @ W1 via fp32 WMMA (one 16-row stripe per block, 8 waves)
    gemm1_wmma_kernel<<<(N + 15) / 16, 256, 0, stream>>>(x, W1, h, N);

    // 3) layer-1 aggregation
    agg1_init_kernel<<<(N * F_HID + 255) / 256, 256, 0, stream>>>(h, dinv, out1, N);
    agg1_edge_kernel<<<(int)((E + 7) / 8), 256, 0, stream>>>(src, dst, dinv, h, out1, E);

    // 4) bias + ReLU + 128->2 projection
    gemm2_kernel<<<(N + 255) / 256, 256, 0, stream>>>(out1, b1, W2, h2, N);

    // 5) layer-2 aggregation straight into d_out (init fully overwrites it)
    agg2_init_kernel<<<(N + 255) / 256, 256, 0, stream>>>(h2, dinv, b2, out, N);
    agg2_edge_kernel<<<(int)((E + 255) / 256), 256, 0, stream>>>(src, dst, dinv, h2, out, E);
}